// Embedding_352187318978
// MI455X (gfx1250) — compile-verified
//
#include <hip/hip_runtime.h>
#include <stdint.h>

// Problem constants (from reference: x[8,2048], weight[1024,32000], bias[1024])
#define D_MODEL 1024
#define VOCAB   32000
#define SEQ     2048
#define BATCH   8
#define PAD_ID  2

typedef float v4f __attribute__((ext_vector_type(4)));

// ---- gfx1250 async global->LDS copy helpers (ASYNCcnt path) ----------------
// LDS byte offset: cast generic pointer -> addrspace(3) (32-bit LDS offset),
// then ptrtoint.
__device__ __forceinline__ uint32_t lds_byte_offset(const void* p) {
  return (uint32_t)(uintptr_t)(__attribute__((address_space(3))) const void*)p;
}

// global_load_async_to_lds_b128: VDST = LDS byte-address VGPR, VADDR = 64-bit
// global address VGPR pair, saddr = off. Tracked with ASYNCcnt.
__device__ __forceinline__ void async_g2lds_b128(uint32_t lds_off, const void* gaddr) {
  asm volatile("global_load_async_to_lds_b128 %0, %1, off"
               :: "v"(lds_off), "v"(gaddr)
               : "memory");
}

__device__ __forceinline__ void wait_asynccnt0() {
  asm volatile("s_wait_asynccnt 0" ::: "memory");
}

// ---------------------------------------------------------------------------
// emb[b,s,d] = weight[d, x[b,s]] + bias[d] + pe[s,d]
// One block per token (b*SEQ+s). 256 threads; thread t owns d = 4t..4t+3 and
// emits a single coalesced 16B non-temporal store.
// ---------------------------------------------------------------------------
__global__ __launch_bounds__(256) void emb_pe_kernel(
    const int* __restrict__ x, const float* __restrict__ w,
    const float* __restrict__ bias, float* __restrict__ out) {
  __shared__ float s_bias[D_MODEL];

  const int tid = threadIdx.x;

  // Stage bias (4KB) into LDS with the async DMA path: 256 lanes x 16B.
  async_g2lds_b128(lds_byte_offset(&s_bias[tid * 4]), bias + tid * 4);
  wait_asynccnt0();
  __syncthreads();

  const int token_idx = blockIdx.x;          // b*SEQ + s
  const int s = token_idx & (SEQ - 1);
  const int t = x[token_idx];                // uniform per block -> scalar load
  const float pos = (float)(s + 1);

  const int d0 = tid * 4;
  const float LOG2_10000 = 13.2877123795494f; // log2(10000)

  v4f v;
#pragma unroll
  for (int k = 0; k < 4; ++k) {
    const int d = d0 + k;
    // Gather: strided column read; weight (128MB) is L2-resident (192MB L2).
    const float wv = w[(size_t)d * VOCAB + t];
    // pe[s,d]: even d -> cos(pos * 10000^(-d/D)), odd d -> sin(pos * 10000^(-(d+1)/D))
    const int ii = (d & 1) ? (d + 1) : d;
    const float e = (float)ii * (1.0f / (float)D_MODEL);
    const float inv_div = exp2f(-e * LOG2_10000);       // v_exp_f32 (native exp2)
    const float angle = pos * inv_div;
    const float pe = (d & 1) ? __sinf(angle) : __cosf(angle); // v_sin/v_cos
    v[k] = wv + s_bias[d] + pe;
  }
  // Streaming output: keep it out of L2 so the weight gather stays resident.
  __builtin_nontemporal_store(v, (v4f*)(out + (size_t)token_idx * D_MODEL + d0));
}

// ---------------------------------------------------------------------------
// input_mask[b,i,j] = (x[b,j] == PAD_ID) ? 1.0f : 0.0f   (broadcast over i)
// Block = (b, 8 rows of i). Async-copy the 8KB x-row into LDS, convert once
// to a float row, then stream it out 8 times with b128 NT stores.
// ---------------------------------------------------------------------------
#define MASK_ROWS 8

__global__ __launch_bounds__(256) void mask_kernel(
    const int* __restrict__ x, float* __restrict__ outm) {
  __shared__ int   s_x[SEQ];   // 8KB
  __shared__ float s_p[SEQ];   // 8KB

  const int tid = threadIdx.x;
  const int blocks_per_b = SEQ / MASK_ROWS;             // 256
  const int b    = blockIdx.x / blocks_per_b;
  const int row0 = (blockIdx.x % blocks_per_b) * MASK_ROWS;

  // Async DMA of x[b, 0..2047] (8KB) into LDS: 256 lanes x 2 x b128.
  const int* xrow = x + b * SEQ;
#pragma unroll
  for (int k = 0; k < 2; ++k) {
    const int j = tid + k * 256;                        // int4 chunk id, 0..511
    async_g2lds_b128(lds_byte_offset(&s_x[j * 4]), xrow + j * 4);
  }
  wait_asynccnt0();
  __syncthreads();

  // Convert to float mask row once (reused by 8 output rows).
#pragma unroll
  for (int k = 0; k < SEQ / 256; ++k) {
    const int j = tid + k * 256;
    s_p[j] = (s_x[j] == PAD_ID) ? 1.0f : 0.0f;
  }
  __syncthreads();

  float* dst = outm + ((size_t)b * SEQ + row0) * SEQ;
#pragma unroll
  for (int r = 0; r < MASK_ROWS; ++r) {
#pragma unroll
    for (int k = 0; k < SEQ / (256 * 4); ++k) {
      const int j4 = tid + k * 256;                     // float4 chunk id
      const v4f v = *(const v4f*)&s_p[j4 * 4];          // ds_load_b128
      __builtin_nontemporal_store(
          v, (v4f*)(dst + (size_t)r * SEQ + j4 * 4));
    }
  }
}

// ---------------------------------------------------------------------------
extern "C" void kernel_launch(void* const* d_in, const int* in_sizes, int n_in,
                              void* d_out, int out_size, void* d_ws, size_t ws_size,
                              hipStream_t stream) {
  (void)in_sizes; (void)n_in; (void)d_ws; (void)ws_size; (void)out_size;

  const int*   x    = (const int*)d_in[0];    // [8, 2048] int32
  const float* w    = (const float*)d_in[1];  // [1024, 32000] f32
  const float* bias = (const float*)d_in[2];  // [1024] f32

  float* out_emb  = (float*)d_out;                                  // [8,2048,1024]
  float* out_mask = (float*)d_out + (size_t)BATCH * SEQ * D_MODEL;  // [8,2048,2048]

  emb_pe_kernel<<<BATCH * SEQ, 256, 0, stream>>>(x, w, bias, out_emb);
  mask_kernel<<<BATCH * (SEQ / MASK_ROWS), 256, 0, stream>>>(x, out_mask);
}